// Wiener_KPN_SA_50216757625487
// MI455X (gfx1250) — compile-verified
//
#include <hip/hip_runtime.h>
#include <cstdint>
#include <cstddef>

#define H_ 320
#define W_ 320
#define B_ 4
#define KH_ 25
#define PAD_ 12
#define HP_ (H_ + 2 * PAD_)      /* 344 */
#define NPB_ (H_ * W_)           /* 102400 */
#define NTOT_ (B_ * NPB_)        /* 409600 */
#define NPAD_ (B_ * HP_ * HP_)   /* 473344 */
#define TOL_ 1e-6f
#define MAXITER_ 20
#define RESTART_ 10

typedef __attribute__((ext_vector_type(2))) float v2f;
typedef __attribute__((ext_vector_type(8))) float v8f;

__device__ __forceinline__ int refl1(int i, int n) {
  if (i < 0) return -i - 1;
  if (i >= n) return 2 * n - 1 - i;
  return i;
}

// ---------------- padding (symmetric, width 12) ----------------
__global__ void k_pad(const float* __restrict__ x, float* __restrict__ xp) {
  int idx = blockIdx.x * 256 + threadIdx.x;
  if (idx >= NPAD_) return;
  int b = idx / (HP_ * HP_);
  int q = idx - b * (HP_ * HP_);
  int qy = q / HP_, qx = q - qy * HP_;
  int sy = refl1(qy - PAD_, H_);
  int sx = refl1(qx - PAD_, W_);
  xp[idx] = x[(size_t)b * NPB_ + sy * W_ + sx];
}

// ---------------- 25x25 correlation via WMMA f32 16x16x4 ----------------
// One wave computes one 16x16 output tile. Banded-GEMM over a 40-wide row
// window: out[y][x] = sum_j patch[y+dy][j] * k[dy][j-x], accumulated over dy.
__global__ void __launch_bounds__(32)
k_conv25(const float* __restrict__ xp, const float* __restrict__ ker,
         float* __restrict__ out, int flip) {
  __shared__ float sP[40 * 40];
  __shared__ float sK[KH_ * KH_];
  const int lane = threadIdx.x;
  const int tx = blockIdx.x, ty = blockIdx.y, b = blockIdx.z;
  const float* xpb = xp + (size_t)b * HP_ * HP_;
  const float* kb = ker + b * KH_ * KH_;
  for (int i = lane; i < KH_ * KH_; i += 32)
    sK[i] = kb[flip ? (KH_ * KH_ - 1 - i) : i];   // reverse both axes
  const int y0 = ty * 16, x0 = tx * 16;           // tile origin (padded coords)
  for (int r0 = 0; r0 < 40; ++r0)
    for (int c = lane; c < 40; c += 32)
      sP[r0 * 40 + c] = xpb[(size_t)(y0 + r0) * HP_ + (x0 + c)];
  __syncthreads();

  const int g = lane >> 4;    // half-wave group: holds K = {2g, 2g+1}
  const int m = lane & 15;    // A-matrix row (M) and B-matrix column (N)
  v8f acc = {};
  for (int dy = 0; dy < KH_; ++dy) {
    const float* krow = &sK[dy * KH_];
    const float* prow = &sP[(m + dy) * 40];
#pragma unroll
    for (int jc = 0; jc < 10; ++jc) {
      const int kidx = jc * 4 + 2 * g;  // K-index of .x component
      v2f a;
      a.x = prow[kidx];
      a.y = prow[kidx + 1];
      int t0 = kidx - m, t1 = kidx + 1 - m;   // band offsets into PSF row
      int t0c = t0 < 0 ? 0 : (t0 > 24 ? 24 : t0);
      int t1c = t1 < 0 ? 0 : (t1 > 24 ? 24 : t1);
      float w0 = krow[t0c];
      float w1 = krow[t1c];
      w0 = ((unsigned)t0 < 25u) ? w0 : 0.f;
      w1 = ((unsigned)t1 < 25u) ? w1 : 0.f;
      v2f bv;
      bv.x = w0;
      bv.y = w1;
      acc = __builtin_amdgcn_wmma_f32_16x16x4_f32(
          false, a, false, bv, (short)0, acc, false, false);
    }
  }
  // D layout: VGPR v -> row M=v (lanes 0-15) / M=v+8 (lanes 16-31), col = lane&15
  float* ob = out + (size_t)b * NPB_;
#pragma unroll
  for (int v = 0; v < 8; ++v)
    ob[(size_t)(ty * 16 + v + 8 * g) * W_ + (tx * 16 + m)] = acc[v];
}

// ---------------- per-pixel filter L1-normalization ----------------
__global__ void k_normf(const float* __restrict__ f, float* __restrict__ fn) {
  int idx = blockIdx.x * 256 + threadIdx.x;
  if (idx >= NTOT_) return;
  int b = idx / NPB_;
  int p = idx - b * NPB_;
  float s = 0.f;
#pragma unroll
  for (int t = 0; t < 9; ++t) s += fabsf(f[((size_t)b * 9 + t) * NPB_ + p]);
  float d = fmaxf(s, 1e-12f);
#pragma unroll
  for (int t = 0; t < 9; ++t) {
    size_t o = ((size_t)b * 9 + t) * NPB_ + p;
    fn[o] = f[o] / d;
  }
}

__global__ void k_rw(const float* __restrict__ reg, float* __restrict__ scal) {
  if (threadIdx.x == 0 && blockIdx.x == 0) scal[0] = expf(reg[0]);
}

__global__ void k_copy(const float* __restrict__ src, float* __restrict__ dst, int n) {
  int i = blockIdx.x * 256 + threadIdx.x;
  if (i < n) dst[i] = src[i];
}

// ---------------- L x : per-pixel 3x3 adaptive conv ----------------
__global__ void k_sconv(const float* __restrict__ fn, const float* __restrict__ x,
                        float* __restrict__ lx) {
  int idx = blockIdx.x * 256 + threadIdx.x;
  if (idx >= NTOT_) return;
  int b = idx / NPB_;
  int p = idx - b * NPB_;
  int y = p / W_, xx = p - y * W_;
  const float* xb = x + (size_t)b * NPB_;
  float acc = 0.f;
#pragma unroll
  for (int i = 0; i < 3; ++i) {
    int ry = refl1(y + i - 1, H_);
#pragma unroll
    for (int j = 0; j < 3; ++j) {
      int rx = refl1(xx + j - 1, W_);
      acc += fn[((size_t)b * 9 + i * 3 + j) * NPB_ + p] * xb[ry * W_ + rx];
    }
  }
  lx[idx] = acc;
}

// ---------------- Ap = KtKx + rw * L^T(lx)  (adjoint incl. pad fold) ----------
__global__ void k_sconv_adj(const float* __restrict__ fn, const float* __restrict__ lx,
                            const float* __restrict__ ktk, const float* __restrict__ scal,
                            float* __restrict__ ap) {
  int idx = blockIdx.x * 256 + threadIdx.x;
  if (idx >= NTOT_) return;
  int b = idx / NPB_;
  int p = idx - b * NPB_;
  int y = p / W_, xx = p - y * W_;
  const float* lxb = lx + (size_t)b * NPB_;
  float acc = 0.f;
  for (int dy = -1; dy <= 1; ++dy) {
    int qy = y + dy;
    if (qy < 0 || qy >= H_) continue;
    for (int dx = -1; dx <= 1; ++dx) {
      int qx = xx + dx;
      if (qx < 0 || qx >= W_) continue;
      float w = 0.f;
#pragma unroll
      for (int i = 0; i < 3; ++i) {
        if (refl1(qy + i - 1, H_) != y) continue;
#pragma unroll
        for (int j = 0; j < 3; ++j) {
          if (refl1(qx + j - 1, W_) != xx) continue;
          w += fn[((size_t)b * 9 + i * 3 + j) * NPB_ + qy * W_ + qx];
        }
      }
      acc += w * lxb[qy * W_ + qx];
    }
  }
  ap[idx] = ktk[idx] + scal[0] * acc;
}

// ---------------- reductions & CG scalar ops ----------------
__device__ float blk_reduce(float v) {
  __shared__ float sm[256];
  int t = threadIdx.x;
  sm[t] = v;
  __syncthreads();
  for (int s = 128; s > 0; s >>= 1) {
    if (t < s) sm[t] += sm[t + s];
    __syncthreads();
  }
  float r = sm[0];
  __syncthreads();
  return r;
}

__global__ void k_dot(const float* __restrict__ u, const float* __restrict__ v,
                      float* __restrict__ out) {
  int b = blockIdx.x;
  const float* ub = u + (size_t)b * NPB_;
  const float* vb = v + (size_t)b * NPB_;
  float acc = 0.f;
  for (int i = threadIdx.x; i < NPB_; i += 256) acc += ub[i] * vb[i];
  float tot = blk_reduce(acc);
  if (threadIdx.x == 0) out[b] = tot;
}

__global__ void k_init_res(const float* __restrict__ bvec, const float* __restrict__ ax,
                           float* __restrict__ r, float* __restrict__ p,
                           float* __restrict__ rs, float* __restrict__ bn2) {
  int b = blockIdx.x;
  size_t off = (size_t)b * NPB_;
  float accR = 0.f, accB = 0.f;
  for (int i = threadIdx.x; i < NPB_; i += 256) {
    float bv = bvec[off + i];
    float rv = bv - ax[off + i];
    r[off + i] = rv;
    p[off + i] = rv;
    accR += rv * rv;
    accB += bv * bv;
  }
  float tr = blk_reduce(accR);
  float tb = blk_reduce(accB);
  if (threadIdx.x == 0) {
    rs[b] = tr;
    bn2[b] = tb;
  }
}

__global__ void k_update(float* __restrict__ x, float* __restrict__ r,
                         const float* __restrict__ p, const float* __restrict__ ap,
                         const float* __restrict__ rs_old, const float* __restrict__ denom,
                         const float* __restrict__ bn2, float* __restrict__ rs_new) {
  int b = blockIdx.x;
  float rsb = rs_old[b];
  bool done = rsb <= (TOL_ * TOL_) * bn2[b];
  float dn = denom[b];
  float alpha = done ? 0.f : rsb / (dn == 0.f ? 1.f : dn);
  size_t off = (size_t)b * NPB_;
  float acc = 0.f;
  for (int i = threadIdx.x; i < NPB_; i += 256) {
    float pi = p[off + i], api = ap[off + i];
    x[off + i] += alpha * pi;
    float rn = r[off + i] - alpha * api;
    r[off + i] = rn;
    acc += rn * rn;
  }
  float t = blk_reduce(acc);
  if (threadIdx.x == 0) rs_new[b] = t;
}

__global__ void k_pupdate(float* __restrict__ p, const float* __restrict__ r,
                          const float* __restrict__ rs_old, const float* __restrict__ rs_new,
                          const float* __restrict__ bn2) {
  int idx = blockIdx.x * 256 + threadIdx.x;
  if (idx >= NTOT_) return;
  int b = idx / NPB_;
  float rsb = rs_old[b];
  bool done = rsb <= (TOL_ * TOL_) * bn2[b];
  if (!done) {
    float beta = rs_new[b] / (rsb == 0.f ? 1.f : rsb);
    p[idx] = r[idx] + beta * p[idx];
  }
}

// ---------------- host-side operator application ----------------
static void Aop(const float* in, float* out, const float* ker, const float* fn,
                float* xp, float* t1, float* ktk, float* lx, const float* scal,
                hipStream_t stream) {
  dim3 cgrid(W_ / 16, H_ / 16, B_);
  int padBlocks = (NPAD_ + 255) / 256;
  int nBlocks = (NTOT_ + 255) / 256;
  k_pad<<<padBlocks, 256, 0, stream>>>(in, xp);
  k_conv25<<<cgrid, 32, 0, stream>>>(xp, ker, t1, 1);   // K x   (flipped)
  k_pad<<<padBlocks, 256, 0, stream>>>(t1, xp);
  k_conv25<<<cgrid, 32, 0, stream>>>(xp, ker, ktk, 0);  // K^T (K x)
  k_sconv<<<nBlocks, 256, 0, stream>>>(fn, in, lx);
  k_sconv_adj<<<nBlocks, 256, 0, stream>>>(fn, lx, ktk, scal, out);
}

extern "C" void kernel_launch(void* const* d_in, const int* in_sizes, int n_in,
                              void* d_out, int out_size, void* d_ws, size_t ws_size,
                              hipStream_t stream) {
  (void)in_sizes; (void)n_in; (void)out_size; (void)ws_size;
  const float* y   = (const float*)d_in[0];
  const float* ker = (const float*)d_in[1];
  const float* frw = (const float*)d_in[2];
  const float* reg = (const float*)d_in[3];
  const float* x0  = (const float*)d_in[4];
  float* x = (float*)d_out;

  float* w = (float*)d_ws;
  float* fn   = w; w += (size_t)9 * NTOT_;
  float* xp   = w; w += NPAD_;
  float* t1   = w; w += NTOT_;
  float* ktk  = w; w += NTOT_;
  float* lx   = w; w += NTOT_;
  float* r    = w; w += NTOT_;
  float* p    = w; w += NTOT_;
  float* ap   = w; w += NTOT_;
  float* bvec = w; w += NTOT_;
  float* scal = w; w += 64;
  float* rs0 = scal + 8;
  float* rs1 = scal + 16;
  float* denom = scal + 24;
  float* bn2 = scal + 32;

  int nBlocks = (NTOT_ + 255) / 256;
  int padBlocks = (NPAD_ + 255) / 256;

  k_normf<<<nBlocks, 256, 0, stream>>>(frw, fn);
  k_rw<<<1, 32, 0, stream>>>(reg, scal);
  k_copy<<<nBlocks, 256, 0, stream>>>(x0, x, NTOT_);

  // b = K^T y
  k_pad<<<padBlocks, 256, 0, stream>>>(y, xp);
  k_conv25<<<dim3(W_ / 16, H_ / 16, B_), 32, 0, stream>>>(xp, ker, bvec, 0);

  // r = b - A x0, p = r, rs0, bnorm^2
  Aop(x, ap, ker, fn, xp, t1, ktk, lx, scal, stream);
  k_init_res<<<B_, 256, 0, stream>>>(bvec, ap, r, p, rs0, bn2);

  for (int i = 0; i < MAXITER_; ++i) {
    float* rsA = (i & 1) ? rs1 : rs0;   // rs entering this iteration
    float* rsB = (i & 1) ? rs0 : rs1;   // rs after r-update
    if (i > 0 && (i % RESTART_) == 0) {
      Aop(x, ap, ker, fn, xp, t1, ktk, lx, scal, stream);
      k_init_res<<<B_, 256, 0, stream>>>(bvec, ap, r, p, rsA, bn2);
    }
    Aop(p, ap, ker, fn, xp, t1, ktk, lx, scal, stream);
    k_dot<<<B_, 256, 0, stream>>>(p, ap, denom);
    k_update<<<B_, 256, 0, stream>>>(x, r, p, ap, rsA, denom, bn2, rsB);
    k_pupdate<<<nBlocks, 256, 0, stream>>>(p, r, rsA, rsB, bn2);
  }
}